// graph_loss_59210419143349
// MI455X (gfx1250) — compile-verified
//
#include <hip/hip_runtime.h>
#include <math.h>

#define D  2048
#define BB 2048
#define TD 6144   // 3*D

typedef float v2f __attribute__((ext_vector_type(2)));
typedef float v8f __attribute__((ext_vector_type(8)));

// ---- workspace layout (float offsets) ----
#define OFF_PART   0                        // 8 matrices * 8 rowchunks * D
#define OFF_COLSUM (OFF_PART + 8*8*D)       // 8 * D
#define OFF_H      (OFF_COLSUM + 8*D)       // 16 * D   (rows 0-3: h0 per call, 4-7: ctx, 8-15: zero pad)
#define OFF_S      (OFF_H + 16*D)           // D
#define OFF_AGG    (OFF_S + D)              // 16 * D   (row 0 = agg0, rest zero pad)
#define OFF_GOUT   (OFF_AGG + 16*D)         // 32 * TD  (rows 0-15: GH, rows 16-31: GI0)
#define OFF_LOSS   (OFF_GOUT + 32*TD)       // 8

// ======================= stage 1: column sums =======================
__global__ void __launch_bounds__(256) colsum_pass1(
    const float* __restrict__ p0, const float* __restrict__ p1,
    const float* __restrict__ p2, const float* __restrict__ p3,
    const float* __restrict__ p4, const float* __restrict__ p5,
    const float* __restrict__ p6, const float* __restrict__ p7,
    float* __restrict__ part) {
  int m = blockIdx.x >> 6;
  const float* src;
  switch (m) {
    case 0: src = p0; break; case 1: src = p1; break;
    case 2: src = p2; break; case 3: src = p3; break;
    case 4: src = p4; break; case 5: src = p5; break;
    case 6: src = p6; break; default: src = p7; break;
  }
  int rem = blockIdx.x & 63;
  int rchunk = rem >> 3, cchunk = rem & 7;
  int col = (cchunk << 8) + threadIdx.x;
  const float* base = src + (size_t)(rchunk << 8) * D + col;
  float acc = 0.0f;
  #pragma unroll 8
  for (int r = 0; r < 256; ++r) acc += base[(size_t)r * D];
  part[(size_t)((m << 3) + rchunk) * D + col] = acc;
}

__global__ void __launch_bounds__(256) colsum_pass2(
    const float* __restrict__ part, float* __restrict__ colsum) {
  int m = blockIdx.x >> 3;
  int col = ((blockIdx.x & 7) << 8) + threadIdx.x;
  float acc = 0.0f;
  #pragma unroll
  for (int rc = 0; rc < 8; ++rc) acc += part[(size_t)((m << 3) + rc) * D + col];
  colsum[(size_t)m * D + col] = acc;
}

// H rows 0-3 = now-means, 4-7 = pre-means, 8-15 = 0; AggPad zeroed.
__global__ void __launch_bounds__(256) init_state(
    const float* __restrict__ colsum, float* __restrict__ Hbuf,
    float* __restrict__ AggPad) {
  int idx = blockIdx.x * 256 + threadIdx.x;   // grid covers 2*16*D
  if (idx < 16 * D) {
    Hbuf[idx] = (idx < 8 * D) ? colsum[idx] * (1.0f / (float)BB) : 0.0f;
  } else {
    AggPad[idx - 16 * D] = 0.0f;
  }
}

// ======================= stage 2: GGC layers =======================
__global__ void __launch_bounds__(256) compute_s(
    const float* __restrict__ Hbuf, float* __restrict__ s) {
  int col = blockIdx.x * 256 + threadIdx.x;
  s[col] = Hbuf[4 * D + col] + Hbuf[5 * D + col] + Hbuf[6 * D + col] + Hbuf[7 * D + col];
}

// agg0[n] = sum_k s[k] * Wg[k][n]   (Wg = ggc_w[layer], row-major [D][D])
__global__ void __launch_bounds__(256) agg_matvec(
    const float* __restrict__ s, const float* __restrict__ Wg,
    float* __restrict__ AggPad) {
  __shared__ float sh[256];
  int n = blockIdx.x * 256 + threadIdx.x;
  float acc = 0.0f;
  for (int kc = 0; kc < D; kc += 256) {
    sh[threadIdx.x] = s[kc + threadIdx.x];
    __syncthreads();
    #pragma unroll 8
    for (int kk = 0; kk < 256; ++kk)
      acc += sh[kk] * Wg[(size_t)(kc + kk) * D + n];
    __syncthreads();
  }
  AggPad[n] = acc;  // row 0; rows 1..15 stay zero from init_state
}

// One M=16 WMMA GEMM per weight matrix:
//   waves   0..383: GH  = H      @ w_hh.T  -> Gout rows  0..15
//   waves 384..767: GI0 = AggPad @ w_ih.T  -> Gout rows 16..31
// (x @ W.T)[n] = dot(x, W[n,:]) so B[k][n] = W[n*D + k].
__global__ void __launch_bounds__(256) ggc_gemm_wmma(
    const float* __restrict__ Hbuf, const float* __restrict__ AggPad,
    const float* __restrict__ w_hh, const float* __restrict__ w_ih,
    float* __restrict__ Gout) {
  int wave = (blockIdx.x * 256 + threadIdx.x) >> 5;   // 0..767
  int lane = threadIdx.x & 31;
  int l16  = lane & 15;
  int koff = (lane >= 16) ? 2 : 0;                    // K half per ISA f32 A/B layout
  bool isGH = wave < 384;
  int tile = isGH ? wave : wave - 384;
  const float* W = isGH ? w_hh : w_ih;
  const float* A = isGH ? Hbuf : AggPad;
  int n = tile * 16 + l16;
  const float* arow = A + (size_t)l16 * D;            // A-matrix row m = lane%16
  const float* brow = W + (size_t)n * D;              // B column n = row n of W
  v8f acc = {};
  for (int k0 = 0; k0 < D; k0 += 4) {
    v2f a = *(const v2f*)(arow + k0 + koff);          // {K=koff, K=koff+1}
    v2f b = *(const v2f*)(brow + k0 + koff);
    acc = __builtin_amdgcn_wmma_f32_16x16x4_f32(
        /*neg_a=*/false, a, /*neg_b=*/false, b,
        /*c_mod=*/(short)0, acc, /*reuse_a=*/false, /*reuse_b=*/false);
  }
  int rowbase = ((lane >= 16) ? 8 : 0) + (isGH ? 0 : 16);
  #pragma unroll
  for (int r = 0; r < 8; ++r)                          // C/D: VGPR r -> M = r (+8 for hi lanes)
    Gout[(size_t)(rowbase + r) * TD + tile * 16 + l16] = acc[r];
}

// GRU gate update for the 8 live rows. Rows 0-3 use gi0 (Gout row 16), rows 4-7 use b_ih only.
__global__ void __launch_bounds__(256) gru_update(
    float* __restrict__ Hbuf, const float* __restrict__ Gout,
    const float* __restrict__ b_ih, const float* __restrict__ b_hh) {
  int row = blockIdx.x >> 3;
  int d = ((blockIdx.x & 7) << 8) + threadIdx.x;
  const float* gh = Gout + (size_t)row * TD;
  float gh_r = gh[d]         + b_hh[d];
  float gh_z = gh[D + d]     + b_hh[D + d];
  float gh_n = gh[2 * D + d] + b_hh[2 * D + d];
  float gi_r, gi_z, gi_n;
  if (row < 4) {
    const float* gi = Gout + (size_t)16 * TD;
    gi_r = gi[d] + b_ih[d];
    gi_z = gi[D + d] + b_ih[D + d];
    gi_n = gi[2 * D + d] + b_ih[2 * D + d];
  } else {
    gi_r = b_ih[d]; gi_z = b_ih[D + d]; gi_n = b_ih[2 * D + d];
  }
  float r  = 1.0f / (1.0f + __expf(-(gi_r + gh_r)));
  float z  = 1.0f / (1.0f + __expf(-(gi_z + gh_z)));
  float nn = tanhf(gi_n + r * gh_n);
  float h  = Hbuf[(size_t)row * D + d];
  Hbuf[(size_t)row * D + d] = (1.0f - z) * nn + z * h;
}

// ======================= stage 3: KDE / KL loss =======================
__device__ __forceinline__ unsigned rotl32(unsigned x, int n) {
  return (x << n) | (x >> (32 - n));
}
__device__ void threefry2x32(unsigned k0, unsigned k1, unsigned c0, unsigned c1,
                             unsigned& o0, unsigned& o1) {
  unsigned ks0 = k0, ks1 = k1, ks2 = k0 ^ k1 ^ 0x1BD11BDAu;
  unsigned x0 = c0 + ks0, x1 = c1 + ks1;
  const int ra[4] = {13, 15, 26, 6}, rb[4] = {17, 29, 16, 24};
  #pragma unroll
  for (int i = 0; i < 4; ++i) { x0 += x1; x1 = rotl32(x1, ra[i]); x1 ^= x0; }
  x0 += ks1; x1 += ks2 + 1u;
  #pragma unroll
  for (int i = 0; i < 4; ++i) { x0 += x1; x1 = rotl32(x1, rb[i]); x1 ^= x0; }
  x0 += ks2; x1 += ks0 + 2u;
  #pragma unroll
  for (int i = 0; i < 4; ++i) { x0 += x1; x1 = rotl32(x1, ra[i]); x1 ^= x0; }
  x0 += ks0; x1 += ks1 + 3u;
  #pragma unroll
  for (int i = 0; i < 4; ++i) { x0 += x1; x1 = rotl32(x1, rb[i]); x1 ^= x0; }
  x0 += ks1; x1 += ks2 + 4u;
  #pragma unroll
  for (int i = 0; i < 4; ++i) { x0 += x1; x1 = rotl32(x1, ra[i]); x1 ^= x0; }
  x0 += ks2; x1 += ks0 + 5u;
  o0 = x0; o1 = x1;
}
__device__ __forceinline__ float tf_normal(unsigned k0, unsigned k1, unsigned idx) {
  unsigned b0, b1;
  threefry2x32(k0, k1, 0u, idx, b0, b1);
  float u = (float)(b0 >> 8) * (1.0f / 16777216.0f);  // [0,1)
  float v = 2.0f * u - 1.0f;
  v = fminf(fmaxf(v, -0.9999997f), 0.9999997f);
  return 1.41421356237f * erfinvf(v);
}

__device__ float block_reduce256(float v, float* red) {
  int tid = threadIdx.x;
  red[tid] = v; __syncthreads();
  #pragma unroll
  for (int s = 128; s > 0; s >>= 1) {
    if (tid < s) red[tid] += red[tid + s];
    __syncthreads();
  }
  float r = red[0]; __syncthreads();
  return r;
}

__global__ void __launch_bounds__(256) kde_kl(
    const float* __restrict__ Hbuf, const float* __restrict__ colsum,
    float* __restrict__ loss_out) {
  __shared__ float d1[D], d2[D], red[256];
  int c = blockIdx.x;  // comparison 0..7
  int tid = threadIdx.x;
  int vrow = (c < 4) ? c : c - 4;
  int srow = (c < 4) ? 4 + c : c - 4;
  const float* v1 = Hbuf + (size_t)vrow * D;     // GGC output vector
  const float* v2 = colsum + (size_t)srow * D;   // *.sum(axis=0)

  unsigned f0, f1, a0, a1, b0, b1;
  threefry2x32(0u, 42u, 0u, (unsigned)c, f0, f1);  // fold_in(key(42), c)
  threefry2x32(f0, f1, 0u, 1u, a0, a1);            // split -> k1
  threefry2x32(f0, f1, 2u, 3u, b0, b1);            // split -> k2

  for (int i = tid; i < D; i += 256) {
    d1[i] = v1[i] + 0.01f * tf_normal(a0, a1, (unsigned)i);
    d2[i] = v2[i] + 0.01f * tf_normal(b0, b1, (unsigned)i);
  }
  __syncthreads();

  float s1 = 0, ss1 = 0, s2 = 0, ss2 = 0;
  for (int i = tid; i < D; i += 256) {
    float x = d1[i]; s1 += x; ss1 += x * x;
    float y = d2[i]; s2 += y; ss2 += y * y;
  }
  s1 = block_reduce256(s1, red);  ss1 = block_reduce256(ss1, red);
  s2 = block_reduce256(s2, red);  ss2 = block_reduce256(ss2, red);

  float fac  = powf((float)D * 0.75f, -0.2f);      // Silverman, d=1
  float var1 = (ss1 - s1 * s1 / (float)D) / (float)(D - 1);
  float var2 = (ss2 - s2 * s2 / (float)D) / (float)(D - 1);
  float bw21 = var1 * fac * fac;
  float bw22 = var2 * fac * fac;
  float c1 = -0.5f / bw21, c2 = -0.5f / bw22;
  float n1 = rsqrtf(6.28318530718f * bw21) / (float)D;
  float n2 = rsqrtf(6.28318530718f * bw22) / (float)D;

  float p[8], q[8], P = 0, Q = 0;
  #pragma unroll
  for (int t = 0; t < 8; ++t) {
    int i = tid + 256 * t;
    float x = d1[i];
    float e1 = 0, e2 = 0;
    for (int j = 0; j < D; ++j) {
      float u = x - d1[j]; e1 += __expf(c1 * u * u);
      float w = x - d2[j]; e2 += __expf(c2 * w * w);
    }
    p[t] = e1 * n1; q[t] = e2 * n2;
    P += p[t]; Q += q[t];
  }
  P = block_reduce256(P, red);
  Q = block_reduce256(Q, red);
  float kl = 0;
  #pragma unroll
  for (int t = 0; t < 8; ++t) {
    float pn = p[t] / P, qn = q[t] / Q;
    kl += pn * __logf(pn / qn);
  }
  kl = block_reduce256(kl, red);
  if (tid == 0) loss_out[c] = 0.5f * kl;   // A = 0.5 for both pre and now terms
}

__global__ void final_sum(const float* __restrict__ loss_parts, float* __restrict__ out) {
  if (threadIdx.x == 0 && blockIdx.x == 0) {
    float s = 0;
    for (int i = 0; i < 8; ++i) s += loss_parts[i];
    out[0] = s;
  }
}

// ======================= launcher =======================
extern "C" void kernel_launch(void* const* d_in, const int* in_sizes, int n_in,
                              void* d_out, int out_size, void* d_ws, size_t ws_size,
                              hipStream_t stream) {
  const float* x0 = (const float*)d_in[0];   // pair_now
  const float* x1 = (const float*)d_in[1];   // person_1_now
  const float* x2 = (const float*)d_in[2];   // person_2_now
  const float* x3 = (const float*)d_in[3];   // scene_now
  const float* x4 = (const float*)d_in[4];   // pair_pre
  const float* x5 = (const float*)d_in[5];   // person_1_pre
  const float* x6 = (const float*)d_in[6];   // person_2_pre
  const float* x7 = (const float*)d_in[7];   // scene_pre
  const float* ggc_w = (const float*)d_in[8];
  const float* w_ih  = (const float*)d_in[9];
  const float* w_hh  = (const float*)d_in[10];
  const float* b_ih  = (const float*)d_in[11];
  const float* b_hh  = (const float*)d_in[12];

  float* ws     = (float*)d_ws;
  float* part   = ws + OFF_PART;
  float* colsum = ws + OFF_COLSUM;
  float* Hbuf   = ws + OFF_H;
  float* svec   = ws + OFF_S;
  float* AggPad = ws + OFF_AGG;
  float* Gout   = ws + OFF_GOUT;
  float* lossp  = ws + OFF_LOSS;

  // stage 1: column sums of the 8 inputs (128 MB, bandwidth-bound)
  colsum_pass1<<<512, 256, 0, stream>>>(x0, x1, x2, x3, x4, x5, x6, x7, part);
  colsum_pass2<<<64, 256, 0, stream>>>(part, colsum);
  init_state<<<256, 256, 0, stream>>>(colsum, Hbuf, AggPad);

  // stage 2: three GGC layers (batched M=16 WMMA GEMMs)
  for (int layer = 0; layer < 3; ++layer) {
    compute_s<<<8, 256, 0, stream>>>(Hbuf, svec);
    agg_matvec<<<8, 256, 0, stream>>>(svec, ggc_w + (size_t)layer * D * D, AggPad);
    ggc_gemm_wmma<<<96, 256, 0, stream>>>(Hbuf, AggPad, w_hh, w_ih, Gout);
    gru_update<<<64, 256, 0, stream>>>(Hbuf, Gout, b_ih, b_hh);
  }

  // stage 3: 8 KDE/KL comparisons + final scalar
  kde_kl<<<8, 256, 0, stream>>>(Hbuf, colsum, lossp);
  final_sum<<<1, 32, 0, stream>>>(lossp, (float*)d_out);
}